// RGATModel_59339268161921
// MI455X (gfx1250) — compile-verified
//
#include <hip/hip_runtime.h>

typedef float v2f __attribute__((ext_vector_type(2)));
typedef float v8f __attribute__((ext_vector_type(8)));

// ---------------- utility: zero a float/u32 region ----------------
__global__ __launch_bounds__(256) void k_zero(float* __restrict__ p, int n) {
  int t = blockIdx.x * 256 + threadIdx.x;
  if (t < n) p[t] = 0.0f;
}

// ---------------- fp32 GEMM via V_WMMA_F32_16X16X4_F32 ----------------
// C[Nrows x Ccols] = A[Nrows x K] @ B[K x Ccols], row-major fp32.
// Block = 256 threads = 8 waves; block tile 16 rows x 128 cols; wave tile 16x16.
// A tile staged in LDS with padded stride (K+4) -> bank-conflict-free reads.
// ISA layout (32-bit A 16x4): lanes 0-15 hold K={k,k+1}, lanes 16-31 hold K={k+2,k+3};
// B rows striped across lanes within a VGPR with the same half-wave K split.
__global__ __launch_bounds__(256) void k_gemm_wmma(
    const float* __restrict__ A, const float* __restrict__ B, float* __restrict__ C,
    int K, int Ccols) {
  __shared__ float As[16 * 260];            // supports K up to 256
  const int stride = K + 4;
  const int tid  = threadIdx.x;
  const int lane = tid & 31;
  const int wv   = tid >> 5;
  const int r0   = blockIdx.x * 16;
  const int c0   = blockIdx.y * 128 + wv * 16;

  // cooperative, fully-convergent load of the 16 x K A tile (16*K multiple of 256)
  for (int i = tid; i < 16 * K; i += 256) {
    int rr = i / K, cc = i - rr * K;
    As[rr * stride + cc] = A[(size_t)(r0 + rr) * K + cc];
  }
  __syncthreads();

  const int half = lane >> 4;               // 0: K=k,k+1   1: K=k+2,k+3
  const int l16  = lane & 15;
  const float* Ap = As + l16 * stride + 2 * half;
  const float* Bp = B + c0 + l16;

  v8f acc = {};
  for (int kk = 0; kk < K; kk += 4) {
    v2f a, b;
    a.x = Ap[kk];
    a.y = Ap[kk + 1];
    b.x = Bp[(size_t)(kk + 2 * half) * Ccols];
    b.y = Bp[(size_t)(kk + 2 * half + 1) * Ccols];
    acc = __builtin_amdgcn_wmma_f32_16x16x4_f32(false, a, false, b, (short)0, acc,
                                                false, false);
  }
  // C/D layout: VGPR i -> row r0+i (lanes 0-15) / r0+8+i (lanes 16-31), col c0+l16
  float* Cp = C + (size_t)(r0 + half * 8) * Ccols + c0 + l16;
#pragma unroll
  for (int i = 0; i < 8; i++) Cp[(size_t)i * Ccols] = acc[i];
}

// ---------------- per-(node,head) attention logits ----------------
// el[n,h] = sum_o feat[n,h,o]*al[h,o]; er likewise. One wave per (n,h).
__global__ __launch_bounds__(256) void k_attn(
    const float* __restrict__ feat, const float* __restrict__ al,
    const float* __restrict__ ar, float* __restrict__ el, float* __restrict__ er,
    int O, int Nn) {
  int w = blockIdx.x * 8 + (threadIdx.x >> 5);
  int lane = threadIdx.x & 31;
  int n = w >> 1, h = w & 1;
  if (n >= Nn) return;
  const float* f  = feat + (size_t)n * (2 * O) + h * O;
  const float* pa = al + h * O;
  const float* pr = ar + h * O;
  float sl = 0.f, sr = 0.f;
  for (int o = lane; o < O; o += 32) {
    float fv = f[o];
    sl += fv * pa[o];
    sr += fv * pr[o];
  }
  for (int off = 16; off; off >>= 1) {
    sl += __shfl_down(sl, off, 32);
    sr += __shfl_down(sr, off, 32);
  }
  if (lane == 0) { el[n * 2 + h] = sl; er[n * 2 + h] = sr; }
}

// monotone float <-> unsigned key mapping (for atomic max on floats)
__device__ __forceinline__ unsigned f2key(float x) {
  unsigned b = __float_as_uint(x);
  return (b & 0x80000000u) ? ~b : (b | 0x80000000u);
}
__device__ __forceinline__ float key2f(unsigned k) {
  return __uint_as_float((k & 0x80000000u) ? (k ^ 0x80000000u) : ~k);
}
__device__ __forceinline__ float lrelu(float v) { return v > 0.f ? v : 0.2f * v; }

// ---------------- edge pass 1: segment max of logits ----------------
__global__ __launch_bounds__(256) void k_edge_max(
    const int* __restrict__ src, const int* __restrict__ dst,
    const float* __restrict__ el, const float* __restrict__ er,
    unsigned* __restrict__ mk, int EH) {
  int t = blockIdx.x * 256 + threadIdx.x;
  if (t >= EH) return;
  int e = t >> 1, h = t & 1;
  int s = src[e], d = dst[e];
  float v = lrelu(el[s * 2 + h] + er[d * 2 + h]);
  atomicMax(&mk[d * 2 + h], f2key(v));
}

// ---------------- edge pass 2: segment sum of exp(e - m) ----------------
__global__ __launch_bounds__(256) void k_edge_sum(
    const int* __restrict__ src, const int* __restrict__ dst,
    const float* __restrict__ el, const float* __restrict__ er,
    const unsigned* __restrict__ mk, float* __restrict__ ssum, int EH) {
  int t = blockIdx.x * 256 + threadIdx.x;
  if (t >= EH) return;
  int e = t >> 1, h = t & 1;
  int s = src[e], d = dst[e];
  float v = lrelu(el[s * 2 + h] + er[d * 2 + h]);
  float m = key2f(mk[d * 2 + h]);
  atomicAdd(&ssum[d * 2 + h], __expf(v - m));
}

// ---------------- edge pass 3: aggregate alpha * feat[src] into dst ----------------
// One wave per edge; lanes stream the 2*O feature floats with atomic adds.
__global__ __launch_bounds__(256) void k_edge_agg(
    const int* __restrict__ src, const int* __restrict__ dst,
    const float* __restrict__ el, const float* __restrict__ er,
    const unsigned* __restrict__ mk, const float* __restrict__ ssum,
    const float* __restrict__ feat, float* __restrict__ agg, int O, int E) {
  int e = blockIdx.x * 8 + (threadIdx.x >> 5);
  if (e >= E) return;
  int lane = threadIdx.x & 31;
  int s = src[e], d = dst[e];
  float alpha[2];
#pragma unroll
  for (int h = 0; h < 2; h++) {
    float v = lrelu(el[s * 2 + h] + er[d * 2 + h]);
    float m = key2f(mk[d * 2 + h]);
    float sum = ssum[d * 2 + h];
    alpha[h] = __expf(v - m) / fmaxf(sum, 1e-9f);
  }
  const float* fs = feat + (size_t)s * (2 * O);
  float* ad = agg + (size_t)d * (2 * O);
  for (int i = lane; i < 2 * O; i += 32) {
    float a = (i < O) ? alpha[0] : alpha[1];
    atomicAdd(&ad[i], a * fs[i]);
  }
}

// ---------------- bias + relation-sum + head-mean (+relu) ----------------
__global__ __launch_bounds__(256) void k_combine(
    const float* __restrict__ agg, const float* __restrict__ b0,
    const float* __restrict__ b1, float* __restrict__ out,
    int Oshift, int doRelu, int total) {
  int t = blockIdx.x * 256 + threadIdx.x;
  if (t >= total) return;
  int O = 1 << Oshift;
  int n = t >> Oshift, o = t & (O - 1);
  float v0 = agg[(size_t)n * 2 * O + o] + b0[o] + b1[o];
  float v1 = agg[(size_t)n * 2 * O + O + o] + b0[O + o] + b1[O + o];
  float v = 0.5f * (v0 + v1);
  if (doRelu) v = fmaxf(v, 0.f);
  out[t] = v;
}

extern "C" void kernel_launch(void* const* d_in, const int* in_sizes, int n_in,
                              void* d_out, int out_size, void* d_ws, size_t ws_size,
                              hipStream_t stream) {
  const int D = 128;
  const int N = in_sizes[0] / D;     // 20000
  const int E = in_sizes[1];         // 320000

  const float* x = (const float*)d_in[0];
  const int* srcs[2] = {(const int*)d_in[1], (const int*)d_in[3]};
  const int* dsts[2] = {(const int*)d_in[2], (const int*)d_in[4]};
  const float* W1[2]  = {(const float*)d_in[5],  (const float*)d_in[9]};
  const float* al1[2] = {(const float*)d_in[6],  (const float*)d_in[10]};
  const float* ar1[2] = {(const float*)d_in[7],  (const float*)d_in[11]};
  const float* b1[2]  = {(const float*)d_in[8],  (const float*)d_in[12]};
  const float* W2[2]  = {(const float*)d_in[13], (const float*)d_in[17]};
  const float* al2[2] = {(const float*)d_in[14], (const float*)d_in[18]};
  const float* ar2[2] = {(const float*)d_in[15], (const float*)d_in[19]};
  const float* b2[2]  = {(const float*)d_in[16], (const float*)d_in[20]};

  // workspace layout (floats): FEAT | AGG | H1 | EL | ER | MK | S  (~103 MB)
  float* FEAT = (float*)d_ws;
  float* AGG  = FEAT + (size_t)N * 512;
  float* H1   = AGG  + (size_t)N * 512;
  float* EL   = H1   + (size_t)N * 256;
  float* ER   = EL   + (size_t)N * 2;
  unsigned* MK = (unsigned*)(ER + (size_t)N * 2);
  float* S    = (float*)(MK + (size_t)N * 2);

  const int EH = E * 2;
  const int ehBlocks   = (EH + 255) / 256;
  const int aggBlocks  = (E + 7) / 8;
  const int attnBlocks = (N * 2 + 7) / 8;

  // ---------------- layer 1: in=128, O=256 per head ----------------
  k_zero<<<(N * 512 + 255) / 256, 256, 0, stream>>>(AGG, N * 512);
  for (int r = 0; r < 2; r++) {
    dim3 g(N / 16, 512 / 128);
    k_gemm_wmma<<<g, 256, 0, stream>>>(x, W1[r], FEAT, 128, 512);
    k_attn<<<attnBlocks, 256, 0, stream>>>(FEAT, al1[r], ar1[r], EL, ER, 256, N);
    k_zero<<<(N * 4 + 255) / 256, 256, 0, stream>>>((float*)MK, N * 4); // MK + S
    k_edge_max<<<ehBlocks, 256, 0, stream>>>(srcs[r], dsts[r], EL, ER, MK, EH);
    k_edge_sum<<<ehBlocks, 256, 0, stream>>>(srcs[r], dsts[r], EL, ER, MK, S, EH);
    k_edge_agg<<<aggBlocks, 256, 0, stream>>>(srcs[r], dsts[r], EL, ER, MK, S,
                                              FEAT, AGG, 256, E);
  }
  k_combine<<<(N * 256 + 255) / 256, 256, 0, stream>>>(AGG, b1[0], b1[1], H1,
                                                       8, 1, N * 256);

  // ---------------- layer 2: in=256, O=128 per head ----------------
  k_zero<<<(N * 256 + 255) / 256, 256, 0, stream>>>(AGG, N * 256);
  for (int r = 0; r < 2; r++) {
    dim3 g(N / 16, 256 / 128);
    k_gemm_wmma<<<g, 256, 0, stream>>>(H1, W2[r], FEAT, 256, 256);
    k_attn<<<attnBlocks, 256, 0, stream>>>(FEAT, al2[r], ar2[r], EL, ER, 128, N);
    k_zero<<<(N * 4 + 255) / 256, 256, 0, stream>>>((float*)MK, N * 4);
    k_edge_max<<<ehBlocks, 256, 0, stream>>>(srcs[r], dsts[r], EL, ER, MK, EH);
    k_edge_sum<<<ehBlocks, 256, 0, stream>>>(srcs[r], dsts[r], EL, ER, MK, S, EH);
    k_edge_agg<<<aggBlocks, 256, 0, stream>>>(srcs[r], dsts[r], EL, ER, MK, S,
                                              FEAT, AGG, 128, E);
  }
  k_combine<<<(N * 128 + 255) / 256, 256, 0, stream>>>(AGG, b2[0], b2[1],
                                                       (float*)d_out, 7, 0, N * 128);
}